// FullyConvolutionalFractionalScaling2D_74629351735526
// MI455X (gfx1250) — compile-verified
//
#include <hip/hip_runtime.h>

typedef float v4f __attribute__((ext_vector_type(4)));

#define HS    256   // coarse rows  (H/s)
#define WS    256   // coarse cols  (W/s)
#define HIN   512
#define WIN   512
#define CH    32
#define BATCH 4
#define HOUT  768
#define WOUT  768

#define WTILE 64              // coarse cells per block along W
#define PIX   (2*WTILE + 2)   // 130 staged input pixels per row
#define NROW  3               // staged input rows
#define NF4   (NROW * PIX * (CH/4))   // 3120 16-byte chunks to stage

__device__ __forceinline__ void compute_cell(const float* __restrict__ smem,
                                             const float w81[81],
                                             float* __restrict__ out,
                                             int b, int hs, int ws0,
                                             int item)
{
    const int wl = item >> 3;          // coarse cell within tile
    const int c4 = (item & 7) * 4;     // channel start

    // 3x3 input taps for this cell (float4 over channels) from LDS
    v4f x[3][3];
#pragma unroll
    for (int p = 0; p < 3; ++p)
#pragma unroll
        for (int q = 0; q < 3; ++q)
            x[p][q] = *(const v4f*)&smem[(p * PIX + (wl * 2 + q)) * CH + c4];

    const int wsg = ws0 + wl;
#pragma unroll
    for (int a = 0; a < 3; ++a) {
#pragma unroll
        for (int b2 = 0; b2 < 3; ++b2) {
            const float* w9 = &w81[(a * 3 + b2) * 9];
            v4f acc = {0.f, 0.f, 0.f, 0.f};
#pragma unroll
            for (int p = 0; p < 3; ++p)
#pragma unroll
                for (int q = 0; q < 3; ++q)
                    acc += w9[p * 3 + q] * x[p][q];
            // out[b, 3hs+a, 3ws+b2, c4..c4+3]; write-once -> non-temporal B128
            const size_t o =
                ((((size_t)b * HOUT + (hs * 3 + a)) * WOUT) + (wsg * 3 + b2)) * CH + c4;
            __builtin_nontemporal_store(acc, (v4f*)(out + o));
        }
    }
}

__global__ __launch_bounds__(256)
void frac_upsample_bilinear_3_2(const float* __restrict__ in,
                                const float* __restrict__ wgt,
                                float* __restrict__ out)
{
    __shared__ __align__(16) float smem[NROW * PIX * CH];  // 49,920 B
    __shared__ float wsm[81];

    const int tid = threadIdx.x;
    const int ws0 = blockIdx.x * WTILE;   // first coarse cell in tile
    const int hs  = blockIdx.y;           // coarse row
    const int b   = blockIdx.z;           // batch

    // ---- stage the 9 bilinear phase filters (81 floats) through LDS ----
    if (tid < 81) wsm[tid] = wgt[tid];

    // ---- async-stage input slab: rows 2hs-1..2hs+1, cols 2ws0-1..+128 ----
    // replicate padding realized by clamping source coordinates.
    const int row0 = hs * 2 - 1;
    const int col0 = ws0 * 2 - 1;
    for (int f = tid; f < NF4; f += 256) {
        const int r   = f / (PIX * 8);
        const int rem = f - r * (PIX * 8);
        const int p   = rem >> 3;        // staged pixel within row
        const int c4  = (rem & 7) * 4;   // channel start
        int gi = row0 + r; gi = gi < 0 ? 0 : (gi > HIN - 1 ? HIN - 1 : gi);
        int gc = col0 + p; gc = gc < 0 ? 0 : (gc > WIN - 1 ? WIN - 1 : gc);
        const float* gp = in + ((((size_t)b * HIN + gi) * WIN) + gc) * CH + c4;
        // wave-relative LDS byte offset = low 32 bits of the generic pointer
        unsigned lds_off = (unsigned)(uintptr_t)(&smem[(r * PIX + p) * CH + c4]);
        asm volatile("global_load_async_to_lds_b128 %0, %1, off"
                     :
                     : "v"(lds_off), "v"(gp)
                     : "memory");
    }
    asm volatile("s_wait_asynccnt 0" ::: "memory");
    __syncthreads();

    // ---- hoist the 81 phase weights out of LDS once per thread ----
    float w81[81];
#pragma unroll
    for (int i = 0; i < 81; ++i) w81[i] = wsm[i];

    // ---- compute: 512 (cell, 4-channel) items over 256 threads = 2 each ----
    compute_cell(smem, w81, out, b, hs, ws0, tid);
    compute_cell(smem, w81, out, b, hs, ws0, tid + 256);
}

extern "C" void kernel_launch(void* const* d_in, const int* in_sizes, int n_in,
                              void* d_out, int out_size, void* d_ws, size_t ws_size,
                              hipStream_t stream)
{
    (void)in_sizes; (void)n_in; (void)out_size; (void)d_ws; (void)ws_size;
    const float* in  = (const float*)d_in[0];   // (4,512,512,32) f32
    const float* wgt = (const float*)d_in[1];   // (9,1,3,3) f32
    float* out = (float*)d_out;                 // (4,768,768,32) f32

    dim3 grid(WS / WTILE, HS, BATCH);           // (4, 256, 4)
    frac_upsample_bilinear_3_2<<<grid, 256, 0, stream>>>(in, wgt, out);
}